// MultiHeadAttention_spatial_shared_70866960384613
// MI455X (gfx1250) — compile-verified
//
#include <hip/hip_runtime.h>
#include <cstdint>
#include <cstddef>

// ---------------------------------------------------------------------------
// MultiHeadAttention_spatial for MI455X (gfx1250, wave32).
// Memory-bound (≈212 MB @ 23.3 TB/s ≈ 9.1 us floor, ~1.3 GFLOP total), so:
//  * fp32 everywhere (compute is free; keeps reference precision)
//  * coalesced float4 NT staging of q/k/v into LDS
//  * NT streaming of the 118 MB distance tensor
//  * V_WMMA_F32_16X16X4_F32 for the 45x45 output projection (exact fp32 WMMA)
// ---------------------------------------------------------------------------

typedef float v2f __attribute__((ext_vector_type(2)));
typedef float v4f __attribute__((ext_vector_type(4)));
typedef float v8f __attribute__((ext_vector_type(8)));

#define BSZ   32
#define SEQ   4096
#define NTOK  (BSZ * SEQ)          // 131072 tokens
#define NPT   15                   // points per token
#define FEAT  45                   // 3 * 15
#define D2    (NPT * NPT)          // 225 distance elems / token
#define TPB   128                  // tokens per block == threads per block
#define CCS   50                   // padded concat row stride (bank-conflict free)

// LDS partition (float offsets)
#define OFF_Q   0
#define OFF_K   (TPB * FEAT)
#define OFF_V   (2 * TPB * FEAT)
#define OFF_VH  (3 * TPB * FEAT)
#define OFF_CC  (OFF_VH + TPB * FEAT)
#define OFF_WO  (OFF_CC + TPB * CCS)
#define OFF_BO  (OFF_WO + 48 * 48)
#define SMEM_FLOATS (OFF_BO + 48)   // 31792 floats = 127168 bytes

__global__ __launch_bounds__(TPB) void mhsa_spatial_kernel(
    const float* __restrict__ gq, const float* __restrict__ gk,
    const float* __restrict__ gv, const float* __restrict__ gdist,
    const float* __restrict__ Wq, const float* __restrict__ bq,
    const float* __restrict__ Wk, const float* __restrict__ bk,
    const float* __restrict__ Wv, const float* __restrict__ bv,
    const float* __restrict__ Wo, const float* __restrict__ bo,
    float* __restrict__ out)
{
    extern __shared__ float smem[];
    float* s_q  = smem + OFF_Q;
    float* s_k  = smem + OFF_K;
    float* s_v  = smem + OFF_V;
    float* s_vh = smem + OFF_VH;   // projected v, per token: [h*15 + m]
    float* s_cc = smem + OFF_CC;   // concat rows, padded to 48 cols, stride 50
    float* s_wo = smem + OFF_WO;   // Wo^T zero-padded to 48x48: s_wo[k*48+n] = Wo[n][k]
    float* s_bo = smem + OFF_BO;

    const int tid  = threadIdx.x;
    const int tok0 = blockIdx.x * TPB;

    // ---- Stage q/k/v: contiguous 23040 B per array, 16B aligned -> float4 NT
    {
        const v4f* q4 = (const v4f*)(gq + (size_t)tok0 * FEAT);
        const v4f* k4 = (const v4f*)(gk + (size_t)tok0 * FEAT);
        const v4f* v4 = (const v4f*)(gv + (size_t)tok0 * FEAT);
        v4f* sq4 = (v4f*)s_q;
        v4f* sk4 = (v4f*)s_k;
        v4f* sv4 = (v4f*)s_v;
        for (int j = tid; j < TPB * FEAT / 4; j += TPB) {
            sq4[j] = __builtin_nontemporal_load(q4 + j);
            sk4[j] = __builtin_nontemporal_load(k4 + j);
            sv4[j] = __builtin_nontemporal_load(v4 + j);
        }
    }
    // ---- Stage Wo^T (zero padded 48x48) and bias (tiny, L2-resident)
    for (int j = tid; j < 48 * 48; j += TPB) {
        const int kk = j / 48, nn = j - kk * 48;
        s_wo[j] = (kk < FEAT && nn < FEAT) ? Wo[nn * FEAT + kk] : 0.0f;
    }
    if (tid < 48) s_bo[tid] = (tid < FEAT) ? bo[tid] : 0.0f;
    __syncthreads();

    // ---- Phase 1: per-token projections + |q k^T| softmax(225) + V reduce
    {
        const float* xq = s_q + tid * FEAT;
        const float* xk = s_k + tid * FEAT;
        const float* xv = s_v + tid * FEAT;
        float* vrow = s_vh + tid * FEAT;
        float* crow = s_cc + tid * CCS;

        #pragma unroll
        for (int h = 0; h < 3; ++h) {
            const float wq0 = Wq[h * 3 + 0], wq1 = Wq[h * 3 + 1], wq2 = Wq[h * 3 + 2];
            const float wk0 = Wk[h * 3 + 0], wk1 = Wk[h * 3 + 1], wk2 = Wk[h * 3 + 2];
            const float wv0 = Wv[h * 3 + 0], wv1 = Wv[h * 3 + 1], wv2 = Wv[h * 3 + 2];
            const float bqh = bq[h], bkh = bk[h], bvh = bv[h];

            float eq[NPT], ek[NPT], vh[NPT];
            float mq = 0.0f, mk = 0.0f;
            #pragma unroll
            for (int n = 0; n < NPT; ++n) {
                // xr[n][d] = x[d*15+n];  qh[h][n] = sum_d xr[n][d]*W[h][d] + b[h]
                float a = fabsf(fmaf(wq2, xq[30 + n], fmaf(wq1, xq[15 + n], fmaf(wq0, xq[n], bqh))));
                eq[n] = a; mq = fmaxf(mq, a);
                float c = fabsf(fmaf(wk2, xk[30 + n], fmaf(wk1, xk[15 + n], fmaf(wk0, xk[n], bkh))));
                ek[n] = c; mk = fmaxf(mk, c);
                float vv = fmaf(wv2, xv[30 + n], fmaf(wv1, xv[15 + n], fmaf(wv0, xv[n], bvh)));
                vh[n] = vv;
                vrow[h * NPT + n] = vv;
            }
            // max over 225 of |q_n|*|k_m| factors: max|q| * max|k|
            const float Mx = mq * mk;
            float ssum = 0.0f;
            float acc[NPT];
            for (int n = 0; n < NPT; ++n) {
                const float qn = eq[n];
                float an = 0.0f;
                #pragma unroll
                for (int m = 0; m < NPT; ++m) {
                    const float e = __expf(fmaf(qn, ek[m], -Mx));
                    ssum += e;
                    an = fmaf(e, vh[m], an);
                }
                acc[n] = an;
            }
            const float inv = 1.0f / ssum;
            #pragma unroll
            for (int n = 0; n < NPT; ++n) crow[n * 3 + h] = acc[n] * inv;
        }
        crow[45] = 0.0f; crow[46] = 0.0f; crow[47] = 0.0f;   // K-pad for WMMA
    }
    __syncthreads();

    // ---- Phase 2: distance bias, streamed NT from global.
    // concat[t][n*3+h] += 2 * sum_m D[t][n][m] * vh[t][h][m]
    // pair p = t*15+n; block walks pairs sequentially -> sequential HBM stream.
    {
        const float* Dg = gdist + (size_t)tok0 * D2;
        for (int it = 0; it < NPT; ++it) {
            const int pair = tid + it * TPB;        // [0, 1920)
            const int t2 = pair / NPT;
            const int n  = pair - t2 * NPT;
            const float* dr = Dg + t2 * D2 + n * NPT;
            float d[NPT];
            #pragma unroll
            for (int m = 0; m < NPT; ++m) d[m] = __builtin_nontemporal_load(dr + m);
            const float* vr = s_vh + t2 * FEAT;
            float* cr = s_cc + t2 * CCS + n * 3;
            #pragma unroll
            for (int h = 0; h < 3; ++h) {
                float s = 0.0f;
                #pragma unroll
                for (int m = 0; m < NPT; ++m) s = fmaf(d[m], vr[h * NPT + m], s);
                cr[h] += 2.0f * s;                  // unique owner per entry: race-free
            }
        }
    }
    __syncthreads();

    // ---- Phase 3: output projection via V_WMMA_F32_16X16X4_F32 (exact fp32)
    // D(16 tok x 16 col) = sum over 12 K-steps of A(16x4) * B(4x16) + bias
    // A layout: lanes 0-15 hold {K=k0,k0+1}, lanes 16-31 hold {K=k0+2,k0+3}, M=lane&15
    // B layout: mirrored; C/D: VGPR r -> M = r + (lane>=16)*8, N = lane&15
    {
        const int wave  = tid >> 5;
        const int lane  = tid & 31;
        const int mrow  = lane & 15;
        const int khalf = (lane >> 4) * 2;

        for (int tile = wave * 2; tile < wave * 2 + 2; ++tile) {
            const float* arow = s_cc + (tile * 16 + mrow) * CCS;
            #pragma unroll
            for (int nt = 0; nt < 3; ++nt) {
                const int ncol = nt * 16 + mrow;
                const float bias = s_bo[ncol];
                v8f c = {bias, bias, bias, bias, bias, bias, bias, bias};
                #pragma unroll
                for (int k0 = 0; k0 < 48; k0 += 4) {
                    v2f a = *(const v2f*)(arow + k0 + khalf);       // 8B-aligned
                    v2f b;
                    b.x = s_wo[(k0 + khalf)     * 48 + ncol];
                    b.y = s_wo[(k0 + khalf + 1) * 48 + ncol];
                    c = __builtin_amdgcn_wmma_f32_16x16x4_f32(
                            false, a, false, b, (short)0, c, false, false);
                }
                const int mbase = tile * 16 + ((lane >> 4) << 3);
                if (ncol < FEAT) {
                    #pragma unroll
                    for (int r = 0; r < 8; ++r) {
                        out[(size_t)(tok0 + mbase + r) * FEAT + ncol] = c[r];
                    }
                }
            }
        }
    }
}

__global__ void mhsa_zero_attval(float* __restrict__ p, int n)
{
    const int i = blockIdx.x * blockDim.x + threadIdx.x;
    if (i < n) p[i] = 0.0f;
}

extern "C" void kernel_launch(void* const* d_in, const int* in_sizes, int n_in,
                              void* d_out, int out_size, void* d_ws, size_t ws_size,
                              hipStream_t stream)
{
    (void)in_sizes; (void)n_in; (void)d_ws; (void)ws_size; (void)out_size;
    // setup_inputs order: q k v distances is_test Wq bq Wk bk Wv bv Wo bo
    const float* q    = (const float*)d_in[0];
    const float* k    = (const float*)d_in[1];
    const float* v    = (const float*)d_in[2];
    const float* dist = (const float*)d_in[3];
    const float* Wq   = (const float*)d_in[5];
    const float* bq   = (const float*)d_in[6];
    const float* Wk   = (const float*)d_in[7];
    const float* bk   = (const float*)d_in[8];
    const float* Wv   = (const float*)d_in[9];
    const float* bv   = (const float*)d_in[10];
    const float* Wo   = (const float*)d_in[11];
    const float* bo   = (const float*)d_in[12];
    float* out = (float*)d_out;

    mhsa_spatial_kernel<<<dim3(NTOK / TPB), dim3(TPB),
                          SMEM_FLOATS * sizeof(float), stream>>>(
        q, k, v, dist, Wq, bq, Wk, bk, Wv, bv, Wo, bo, out);

    float* attval = out + (size_t)NTOK * FEAT;
    mhsa_zero_attval<<<dim3((SEQ + 255) / 256), dim3(256), 0, stream>>>(attval, SEQ);
}